// InGramRelationLayer_49744311222486
// MI455X (gfx1250) — compile-verified
//
#include <hip/hip_runtime.h>
#include <hip/hip_bf16.h>

#define NUM_REL   20000
#define NUM_EDGES 640000
#define DIM_IN    128
#define DIM_OUT   256
#define NUM_HEAD  8
#define DIM_HID   32
#define PCOLS     768   // [Zh (256) | Zt (256) | Msg (256)]

typedef float v2f __attribute__((ext_vector_type(2)));
typedef float v8f __attribute__((ext_vector_type(8)));

// -------------------------------------------------------------------------
// Float atomic max via int/uint atomics (exact for all non-NaN floats).
// -------------------------------------------------------------------------
__device__ __forceinline__ void atomicMaxF(float* addr, float val) {
  if (val >= 0.0f)
    atomicMax((int*)addr, __float_as_int(val));
  else
    atomicMin((unsigned int*)addr, (unsigned int)__float_as_int(val));
}

// -------------------------------------------------------------------------
// Init: zero d_out, set amax = -inf, sums = 0.
// -------------------------------------------------------------------------
__global__ void init_kernel(float* __restrict__ out, float* __restrict__ amax,
                            float* __restrict__ sums) {
  int i = blockIdx.x * blockDim.x + threadIdx.x;
  if (i < NUM_REL * DIM_OUT) out[i] = 0.0f;
  if (i < NUM_REL * NUM_HEAD) {
    amax[i] = __int_as_float(0xff800000);  // -inf
    sums[i] = 0.0f;
  }
}

// -------------------------------------------------------------------------
// Fused per-relation projection GEMM via V_WMMA_F32_16X16X4_F32.
//   P[20000 x 768] = emb_rel[20000 x 128] @ [W_h^T | W_t^T | Wa^T]
// One wave per 16x16 output tile. grid = (1250, 48), block = 32.
// -------------------------------------------------------------------------
__global__ __launch_bounds__(32)
void gemm_precompute(const float* __restrict__ emb,      // 20000 x 128
                     const float* __restrict__ attn_w,   // 256 x 256
                     const float* __restrict__ aggr_w,   // 256 x 128
                     float* __restrict__ P)              // 20000 x 768
{
  const int lane = threadIdx.x;     // wave32: 0..31
  const int r    = lane & 15;       // N (for B/C) or M (for A) within tile
  const int half = lane >> 4;       // K-pair selector

  const int mt = blockIdx.x;        // 0..1249  (M tile)
  const int nt = blockIdx.y;        // 0..47    (N tile over 768 cols)
  const int n  = nt * 16 + r;       // global output column

  // Column n of B == row of the corresponding weight matrix (K contiguous).
  const float* wrow;
  if (nt < 16)       wrow = attn_w + n * (2 * DIM_IN);               // W_h: attn_w[n][0:128]
  else if (nt < 32)  wrow = attn_w + (n - 256) * (2 * DIM_IN) + 128; // W_t: attn_w[n][128:256]
  else               wrow = aggr_w + (n - 512) * DIM_IN;             // Wa : aggr_w[n][0:128]

  const float* arow = emb + (size_t)(mt * 16 + r) * DIM_IN;

  v8f c = {};
#pragma unroll
  for (int k0 = 0; k0 < DIM_IN; k0 += 4) {
    const int k = k0 + 2 * half;
    v2f a = *(const v2f*)(arow + k);   // A[r][k], A[r][k+1]
    v2f b = *(const v2f*)(wrow + k);   // B[k][n], B[k+1][n]
    c = __builtin_amdgcn_wmma_f32_16x16x4_f32(
        /*neg_a=*/false, a, /*neg_b=*/false, b,
        /*c_mod=*/(short)0, c, /*reuse_a=*/false, /*reuse_b=*/false);
  }

  // C/D layout: lane l -> row M = v + 8*half, col N = r
  float* prow = P + (size_t)(mt * 16) * PCOLS + nt * 16 + r;
#pragma unroll
  for (int v = 0; v < 8; ++v)
    prow[(size_t)(v + 8 * half) * PCOLS] = c[v];
}

// -------------------------------------------------------------------------
// Pass 1: per-edge attention logit + segment max. One wave per edge.
//   z = Zh[head] + Zt[tail] + b ; a = leaky_relu(z)
//   raw[e,h] = sum_d a[h,d]*attn_vec[h,d] + attn_bin[bin,h]
// Lane l handles elements c = l + 32*i  (head = i, d = l).
// -------------------------------------------------------------------------
__global__ __launch_bounds__(256)
void edge_attn_raw(const int* __restrict__ trip, const float* __restrict__ P,
                   const float* __restrict__ attn_b,
                   const float* __restrict__ attn_bin,   // 10 x 8
                   const float* __restrict__ attn_vec,   // 8 x 32
                   float* __restrict__ raw,              // E x 8
                   float* __restrict__ amax)             // R x 8
{
  const int e = blockIdx.x * (blockDim.x >> 5) + (threadIdx.x >> 5);
  if (e >= NUM_EDGES) return;
  const int lane = threadIdx.x & 31;

  const int h  = trip[e * 3 + 0];
  const int t  = trip[e * 3 + 1];
  const int bn = trip[e * 3 + 2];

  const float* zh = P + (size_t)h * PCOLS;        // cols 0:256
  const float* zt = P + (size_t)t * PCOLS + 256;  // cols 256:512

  float part[NUM_HEAD];
#pragma unroll
  for (int i = 0; i < NUM_HEAD; ++i) {
    const int c = lane + 32 * i;                  // head i, d = lane
    float z = zh[c] + zt[c] + attn_b[c];
    float a = z > 0.0f ? z : 0.2f * z;            // leaky_relu(0.2)
    part[i] = a * attn_vec[c];
  }

  // Butterfly reduce all 8 head-partials across the wave.
#pragma unroll
  for (int off = 16; off > 0; off >>= 1) {
#pragma unroll
    for (int i = 0; i < NUM_HEAD; ++i)
      part[i] += __shfl_xor(part[i], off, 32);
  }

  if (lane < NUM_HEAD) {
    float rv = part[lane] + attn_bin[bn * NUM_HEAD + lane];
    raw[(size_t)e * NUM_HEAD + lane] = rv;
    atomicMaxF(&amax[h * NUM_HEAD + lane], rv);
  }
}

// -------------------------------------------------------------------------
// Pass 2: attn_val = exp(raw - amax[head]); segment sum. One thread per (e,h).
// Overwrites raw with attn_val.
// -------------------------------------------------------------------------
__global__ __launch_bounds__(256)
void edge_softmax_num(const int* __restrict__ trip, const float* __restrict__ amax,
                      float* __restrict__ raw, float* __restrict__ sums)
{
  const int i = blockIdx.x * blockDim.x + threadIdx.x;
  if (i >= NUM_EDGES * NUM_HEAD) return;
  const int e  = i >> 3;
  const int hh = i & 7;
  const int h  = trip[e * 3 + 0];
  float v = expf(raw[i] - amax[h * NUM_HEAD + hh]);
  raw[i] = v;
  atomicAdd(&sums[h * NUM_HEAD + hh], v);
}

// -------------------------------------------------------------------------
// Pass 3: out[head] += beta * (Msg[tail] + aggr_b). One wave per edge.
// -------------------------------------------------------------------------
__global__ __launch_bounds__(256)
void edge_scatter(const int* __restrict__ trip, const float* __restrict__ P,
                  const float* __restrict__ aggr_b,
                  const float* __restrict__ attn_val,  // E x 8
                  const float* __restrict__ sums,      // R x 8
                  float* __restrict__ out)             // R x 256
{
  const int e = blockIdx.x * (blockDim.x >> 5) + (threadIdx.x >> 5);
  if (e >= NUM_EDGES) return;
  const int lane = threadIdx.x & 31;

  const int h = trip[e * 3 + 0];
  const int t = trip[e * 3 + 1];

  const float* msg = P + (size_t)t * PCOLS + 512;  // cols 512:768
  float* orow = out + (size_t)h * DIM_OUT;

#pragma unroll
  for (int i = 0; i < NUM_HEAD; ++i) {
    const int c = lane + 32 * i;                   // head i, d = lane
    float beta = attn_val[(size_t)e * NUM_HEAD + i] /
                 (sums[h * NUM_HEAD + i] + 1e-16f);
    atomicAdd(&orow[c], beta * (msg[c] + aggr_b[c]));
  }
}

// -------------------------------------------------------------------------
extern "C" void kernel_launch(void* const* d_in, const int* in_sizes, int n_in,
                              void* d_out, int out_size, void* d_ws, size_t ws_size,
                              hipStream_t stream) {
  const float* emb      = (const float*)d_in[0];
  const int*   trip     = (const int*)  d_in[1];
  const float* attn_w   = (const float*)d_in[2];
  const float* attn_b   = (const float*)d_in[3];
  const float* attn_bin = (const float*)d_in[4];
  const float* attn_vec = (const float*)d_in[5];
  const float* aggr_w   = (const float*)d_in[6];
  const float* aggr_b   = (const float*)d_in[7];
  float* out = (float*)d_out;

  // Workspace layout:
  //   P    : NUM_REL  * 768 floats  (61.44 MB)
  //   raw  : NUM_EDGES * 8  floats  (20.48 MB)  (becomes attn_val in pass 2)
  //   amax : NUM_REL  * 8   floats  ( 0.64 MB)
  //   sums : NUM_REL  * 8   floats  ( 0.64 MB)
  float* P    = (float*)d_ws;
  float* raw  = P    + (size_t)NUM_REL * PCOLS;
  float* amax = raw  + (size_t)NUM_EDGES * NUM_HEAD;
  float* sums = amax + (size_t)NUM_REL * NUM_HEAD;

  {  // zero out + init segment-max/-sum buffers
    int n = NUM_REL * DIM_OUT;  // 5.12M threads covers everything
    init_kernel<<<(n + 255) / 256, 256, 0, stream>>>(out, amax, sums);
  }
  {  // fused per-relation projection GEMM (WMMA f32)
    dim3 grid(NUM_REL / 16, PCOLS / 16);  // 1250 x 48
    gemm_precompute<<<grid, 32, 0, stream>>>(emb, attn_w, aggr_w, P);
  }
  edge_attn_raw<<<NUM_EDGES / 8, 256, 0, stream>>>(trip, P, attn_b, attn_bin,
                                                   attn_vec, raw, amax);
  edge_softmax_num<<<NUM_EDGES * NUM_HEAD / 256, 256, 0, stream>>>(trip, amax,
                                                                   raw, sums);
  edge_scatter<<<NUM_EDGES / 8, 256, 0, stream>>>(trip, P, aggr_b, raw, sums,
                                                  out);
}